// CombinedLoss_55559696941630
// MI455X (gfx1250) — compile-verified
//
#include <hip/hip_runtime.h>

typedef __attribute__((ext_vector_type(2))) float v2f;
typedef __attribute__((ext_vector_type(8))) float v8f;

#define NROWS 32768
#define W 1024

// db8 decomposition low-pass filter (pywt 'db8' dec_lo)
__device__ __constant__ float c_LO[16] = {
    -0.00011747678400228192f, 0.0006754494059985568f, -0.0003917403729959771f,
    -0.00487035299301066f,     0.008746094047015655f,   0.013981027917015516f,
    -0.04408825393106472f,    -0.01736930100202211f,    0.128747426620186f,
     0.00047248457399797254f, -0.2840155429624281f,    -0.015829105256023893f,
     0.5853546836548691f,      0.6756307362980128f,     0.3128715909144659f,
     0.05441584224308161f};

// Normalizers folded into the accumulators.
constexpr float MSE_SCALE  = 1.0f / 33554432.0f;              // 32*1024*1024
constexpr float SPEC_SCALE = 1.0f / 33488896.0f;              // 32*1024*1022
constexpr float WAV1_SCALE = 1.0f / (3.0f * 32768.0f * 519.0f);
constexpr float WAV2_SCALE = 1.0f / (3.0f * 32768.0f * 267.0f);
constexpr float WAV3_SCALE = 1.0f / (3.0f * 32768.0f * 141.0f);

// Symmetric-pad fill. Buffer layout: buf[i+2] = xe[i]; interior a[t] = buf[16+t].
// Left pad  buf[2..15]      = buf[29-l]      (a[13],...,a[0])
// Right pad buf[L+16..L+30] = buf[L+29-l]    (a[L-1],...,a[L-15])
__device__ __forceinline__ void fill_pads(float* buf, int L, int lane) {
    if (lane < 14)       buf[2 + lane]     = buf[29 - lane];
    else if (lane < 29)  buf[L + 2 + lane] = buf[L + 29 - lane];
}

// One DWT level on one row-pair (x,y), one wave.
// bufX/bufY: padded inputs (xe at [2, L+30]). outX/outY: next-level buffers
// (cA written at [16, 16+outCount)), or nullptr at the last level.
__device__ __forceinline__ void dwt_level(const float* bufX, const float* bufY,
                                          float* outX, float* outY,
                                          int L, float scale, int lane,
                                          const v2f bfrag[4], float& acc) {
    const int n        = lane & 15;
    const int hi       = lane >> 4;
    const int mBase    = hi ? 8 : 0;
    const int outCount = (L + 15) >> 1;
    for (int j0 = 0; j0 < outCount; j0 += 16) {
        v8f cx = {0.f, 0.f, 0.f, 0.f, 0.f, 0.f, 0.f, 0.f};
        v8f cy = {0.f, 0.f, 0.f, 0.f, 0.f, 0.f, 0.f, 0.f};
        // A fragment: lane holds row M=n, K pair {0,1} (lanes<16) or {2,3}.
        const int col = 2 + 2 * (j0 + n) + (hi ? 2 : 0);
#pragma unroll
        for (int kk = 0; kk < 4; ++kk) {
            v2f ax = *(const v2f*)(bufX + col + 4 * kk);
            cx = __builtin_amdgcn_wmma_f32_16x16x4_f32(false, ax, false, bfrag[kk],
                                                       (short)0, cx, false, false);
        }
#pragma unroll
        for (int kk = 0; kk < 4; ++kk) {
            v2f ay = *(const v2f*)(bufY + col + 4 * kk);
            cy = __builtin_amdgcn_wmma_f32_16x16x4_f32(false, ay, false, bfrag[kk],
                                                       (short)0, cy, false, false);
        }
        // Column 1 (lanes 1,17): detail coefficients -> loss accumulation.
        if (n == 1) {
#pragma unroll
            for (int g = 0; g < 8; ++g) {
                int j = j0 + mBase + g;
                if (j < outCount) {
                    float dd = cx[g] - cy[g];
                    acc += dd * dd * scale;
                }
            }
        }
        // Column 0 (lanes 0,16): approximation -> next-level input.
        if (outX != nullptr && n == 0) {
#pragma unroll
            for (int g = 0; g < 8; ++g) {
                int j = j0 + mBase + g;
                if (j < outCount) {
                    outX[16 + j] = cx[g];
                    outY[16 + j] = cy[g];
                }
            }
        }
    }
}

__global__ __launch_bounds__(32) void combined_loss_kernel(
    const float* __restrict__ X, const float* __restrict__ Y,
    float* __restrict__ partial) {
    // Ping-pong padded row buffers. Sizes cover last-tile over-reads:
    // L=1024 reads up to idx 1071 (<1088); L=519 up to 559 (<576); L=267 up to 303.
    __shared__ __align__(16) float sx0[1088];
    __shared__ __align__(16) float sy0[1088];
    __shared__ __align__(16) float sx1[576];
    __shared__ __align__(16) float sy1[576];

    const int row  = blockIdx.x;
    const int lane = threadIdx.x;
    const float* xr = X + (size_t)row * W;
    const float* yr = Y + (size_t)row * W;

    float acc = 0.f;

    // Coalesced float4 load of the row pair + MSE on the fly.
#pragma unroll
    for (int i = 0; i < 8; ++i) {
        int e = (i * 32 + lane) * 4;
        float4 vx = *(const float4*)(xr + e);
        float4 vy = *(const float4*)(yr + e);
        *(float4*)(sx0 + 16 + e) = vx;
        *(float4*)(sy0 + 16 + e) = vy;
        float d0 = vx.x - vy.x, d1 = vx.y - vy.y;
        float d2 = vx.z - vy.z, d3 = vx.w - vy.w;
        acc += (d0 * d0 + d1 * d1 + d2 * d2 + d3 * d3) * MSE_SCALE;
    }
    __syncthreads();
    fill_pads(sx0, 1024, lane);
    fill_pads(sy0, 1024, lane);
    __syncthreads();

    // Spectral: second difference over the output row only.
    for (int i = lane; i < W - 2; i += 32) {
        float s = sx0[16 + i] - 2.f * sx0[16 + i + 1] + sx0[16 + i + 2];
        acc += s * s * SPEC_SCALE;
    }

    // B fragments: column 0 = reversed LO, column 1 = reversed HI
    // (HI[15-k] = (-1)^k * LO[k]); other 14 columns zero.
    const int n  = lane & 15;
    const int hi = lane >> 4;
    v2f bfrag[4];
#pragma unroll
    for (int kk = 0; kk < 4; ++kk) {
        int kb = kk * 4 + (hi ? 2 : 0);
        float b0 = 0.f, b1 = 0.f;
        if (n == 0) {
            b0 = c_LO[15 - kb];
            b1 = c_LO[14 - kb];
        } else if (n == 1) {
            b0 = (kb & 1) ? -c_LO[kb] : c_LO[kb];
            b1 = ((kb + 1) & 1) ? -c_LO[kb + 1] : c_LO[kb + 1];
        }
        bfrag[kk][0] = b0;
        bfrag[kk][1] = b1;
    }

    // Level 1: 1024 -> 519
    dwt_level(sx0, sy0, sx1, sy1, 1024, WAV1_SCALE, lane, bfrag, acc);
    __syncthreads();
    fill_pads(sx1, 519, lane);
    fill_pads(sy1, 519, lane);
    __syncthreads();
    // Level 2: 519 -> 267
    dwt_level(sx1, sy1, sx0, sy0, 519, WAV2_SCALE, lane, bfrag, acc);
    __syncthreads();
    fill_pads(sx0, 267, lane);
    fill_pads(sy0, 267, lane);
    __syncthreads();
    // Level 3: 267 -> 141 (details only)
    dwt_level(sx0, sy0, nullptr, nullptr, 267, WAV3_SCALE, lane, bfrag, acc);

    // Wave32 reduction, one partial per row.
#pragma unroll
    for (int off = 16; off > 0; off >>= 1)
        acc += __shfl_down(acc, off, 32);
    if (lane == 0) partial[row] = acc;
}

__global__ void reduce_kernel(const float* __restrict__ partial,
                              float* __restrict__ out) {
    __shared__ float s[256];
    float a = 0.f;
    for (int i = threadIdx.x; i < NROWS; i += 256) a += partial[i];
    s[threadIdx.x] = a;
    __syncthreads();
    for (int off = 128; off > 0; off >>= 1) {
        if ((int)threadIdx.x < off) s[threadIdx.x] += s[threadIdx.x + off];
        __syncthreads();
    }
    if (threadIdx.x == 0) out[0] = s[0];
}

extern "C" void kernel_launch(void* const* d_in, const int* in_sizes, int n_in,
                              void* d_out, int out_size, void* d_ws, size_t ws_size,
                              hipStream_t stream) {
    (void)in_sizes; (void)n_in; (void)out_size; (void)ws_size;
    const float* X = (const float*)d_in[0];
    const float* Y = (const float*)d_in[1];
    float* partial = (float*)d_ws;   // NROWS floats = 128 KB scratch
    float* out     = (float*)d_out;

    combined_loss_kernel<<<NROWS, 32, 0, stream>>>(X, Y, partial);
    reduce_kernel<<<1, 256, 0, stream>>>(partial, out);
}